// GodelFlow_34016140985088
// MI455X (gfx1250) — compile-verified
//
#include <hip/hip_runtime.h>
#include <stdint.h>

// out = x - sign(x) * s*(1-s) * log(p_col),  s = sigmoid(100*(|x|-0.1))
// [65536 x 1024] f32 streaming kernel: 512MB traffic -> ~22us HBM floor.
// CDNA5 async global->LDS ring pipeline, 4 deep (ASYNCcnt-tracked), so each
// wave keeps ~2KB in flight -> enough aggregate outstanding bytes to cover
// the HBM latency-bandwidth product at 23.3 TB/s.

typedef float v4f __attribute__((ext_vector_type(4)));

#define DIM 1024
#define TPB 256            // 8 wave32s; thread t owns columns 4t..4t+3
#define ROW_BYTES (DIM * 4)
#define DEPTH 4            // async ring depth (power of two)

// Issue a 16-byte async copy: global memory at base plus byte_off to the LDS address in lds_addr
__device__ __forceinline__ void async_load16(unsigned lds_addr, unsigned byte_off,
                                             uint64_t base) {
    asm volatile("global_load_async_to_lds_b128 %0, %1, %2"
                 :
                 : "v"(lds_addr), "v"(byte_off), "s"(base)
                 : "memory");
}

__device__ __forceinline__ float godel_elem(float xv, float lp) {
    // t = 100*(|x|-0.1);  e = exp(-t);  s(1-s) = e/(1+e)^2
    float e   = __expf(__fmaf_rn(-100.0f, fabsf(xv), 10.0f));
    float r   = __builtin_amdgcn_rcpf(1.0f + e);
    float sds = e * r * r;
    float sg  = (xv > 0.0f) ? 1.0f : ((xv < 0.0f) ? -1.0f : 0.0f);
    // out = x - sign(x) * lp * s(1-s)    (0.01 * 100 cancels)
    return __fmaf_rn(-sg, lp * sds, xv);
}

__global__ __launch_bounds__(TPB)
void godel_flow_kernel(const float* __restrict__ x,
                       const float* __restrict__ primes,
                       float* __restrict__ out, int rows) {
    __shared__ float buf[DEPTH][DIM];       // 4 x 4KB row buffers, per-wave disjoint

    const int t   = threadIdx.x;
    const int col = t << 2;                 // this thread's 4 columns (fixed for all rows)

    // Per-column log(primes), computed once, amortized over the grid-stride loop.
    v4f p4 = *(const v4f*)(primes + col);
    v4f lp;
    lp.x = __logf(p4.x);
    lp.y = __logf(p4.y);
    lp.z = __logf(p4.z);
    lp.w = __logf(p4.w);

    const uint64_t xbase  = (uint64_t)(uintptr_t)x;
    const unsigned lane_b = (unsigned)col * 4u;   // byte offset within a row
    unsigned lds_addr[DEPTH];
#pragma unroll
    for (int d = 0; d < DEPTH; ++d)
        lds_addr[d] = (unsigned)(uintptr_t)&buf[d][col];

    const int r0 = (int)blockIdx.x;
    if (r0 >= rows) return;
    const int stride = (int)gridDim.x;

    // Prologue: fill the ring with up to DEPTH row copies.
#pragma unroll
    for (int d = 0; d < DEPTH; ++d) {
        const int rr = r0 + d * stride;
        if (rr < rows)
            async_load16(lds_addr[d], (unsigned)rr * (unsigned)ROW_BYTES + lane_b, xbase);
    }

    int rp  = r0;
    int cur = 0;
    while (rp < rows) {
        // Completions are in-order: with DEPTH-1 prefetches still wanted in
        // flight, waiting for ASYNCcnt<=DEPTH-1 means the oldest row (rp) has
        // landed. In the drain region just wait for everything.
        if (rp + (DEPTH - 1) * stride < rows) {
            asm volatile("s_wait_asynccnt 0x3" ::: "memory");
        } else {
            asm volatile("s_wait_asynccnt 0x0" ::: "memory");
        }

        // Each wave reads back only the LDS bytes it issued itself -> no barrier.
        v4f xv = *(const v4f*)&buf[cur][col];   // ds_load_b128
        v4f o;
        o.x = godel_elem(xv.x, lp.x);
        o.y = godel_elem(xv.y, lp.y);
        o.z = godel_elem(xv.z, lp.z);
        o.w = godel_elem(xv.w, lp.w);

        // Streaming output: never re-read, 256MB > L2 -> non-temporal store.
        __builtin_nontemporal_store(o, (v4f*)(out + (size_t)rp * DIM + col));

        // Refill this ring slot DEPTH rows ahead. Issued after the store (asm
        // memory clobber keeps it there), so the ds_load above has fully
        // completed (s_wait_dscnt precedes the dependent compute) -> no WAR
        // race between the async LDS write and the just-finished LDS read.
        const int ri = rp + DEPTH * stride;
        if (ri < rows)
            async_load16(lds_addr[cur], (unsigned)ri * (unsigned)ROW_BYTES + lane_b, xbase);

        rp += stride;
        cur = (cur + 1) & (DEPTH - 1);
    }
}

extern "C" void kernel_launch(void* const* d_in, const int* in_sizes, int n_in,
                              void* d_out, int out_size, void* d_ws, size_t ws_size,
                              hipStream_t stream) {
    (void)n_in; (void)out_size; (void)d_ws; (void)ws_size;
    const float* x      = (const float*)d_in[0];
    const float* primes = (const float*)d_in[1];
    float*       out    = (float*)d_out;

    const int rows   = in_sizes[0] / DIM;          // 65536
    int       blocks = rows < 4096 ? rows : 4096;  // grid-stride: 16 rows/block
    if (blocks < 1) blocks = 1;

    godel_flow_kernel<<<blocks, TPB, 0, stream>>>(x, primes, out, rows);
}